// Propagator1D_82016695484589
// MI455X (gfx1250) — compile-verified
//
#include <hip/hip_runtime.h>

// ---------------------------------------------------------------------------
// 1-D wave propagator step, MI455X (gfx1250).
// Memory-bound (~540 MB traffic, ~0.4 GFLOP) -> LDS-staged streaming kernel.
// Stencil computed on the matrix pipe via V_WMMA_F32_16X16X4_F32:
//   D[16 z x 16 shot] = sum_{c=0..4} A_c(16x4 band) x U_c(4x16).
// Inputs staged global->LDS via CDNA5 async-to-LDS (b128 where aligned);
// outputs streamed as non-temporal b128 stores.
// ---------------------------------------------------------------------------

namespace {
constexpr int NZv    = 4000;
constexpr int NPADv  = 100;
constexpr int LP     = NZv + NPADv;     // 4100 padded length (multiple of 4)
constexpr int NV     = 128;
constexpr int NS     = 64;
constexpr int ZC     = 384;             // z-chunk per block (multiple of 16)
constexpr int PITCH  = ZC + 4;          // 388 ; %64==4 -> conflict-free, %4==0
constexpr int CPITCH = 452;             // curr pitch: >= ZC+8, %64==4, %4==0
constexpr int NT     = 256;             // 8 wave32
}

typedef __attribute__((ext_vector_type(2))) float v2f;
typedef __attribute__((ext_vector_type(4))) float v4f;
typedef __attribute__((ext_vector_type(8))) float v8f;
typedef __attribute__((ext_vector_type(4))) int   v4i;

// ---- CDNA5 async global->LDS staging (ASYNCcnt path) ----------------------
// b32 signature confirmed by clang diagnostic: (global int*, local int*, Ii, Ii)
__device__ __forceinline__ void ld_async_f32(float* lds_dst, const float* gsrc) {
#if __has_builtin(__builtin_amdgcn_global_load_async_to_lds_b32)
  int* gi = (int*)const_cast<float*>(gsrc);
  int* li = (int*)lds_dst;
  __builtin_amdgcn_global_load_async_to_lds_b32(
      (__attribute__((address_space(1))) int*)gi,
      (__attribute__((address_space(3))) int*)li,
      0 /*imm offset*/, 0 /*cpol*/);
#else
  *lds_dst = *gsrc;
#endif
}

__device__ __forceinline__ void ld_async_f32x4(float* lds_dst, const float* gsrc) {
#if __has_builtin(__builtin_amdgcn_global_load_async_to_lds_b128)
  v4i* gi = (v4i*)const_cast<float*>(gsrc);
  v4i* li = (v4i*)lds_dst;
  __builtin_amdgcn_global_load_async_to_lds_b128(
      (__attribute__((address_space(1))) v4i*)gi,
      (__attribute__((address_space(3))) v4i*)li,
      0 /*imm offset*/, 0 /*cpol*/);
#else
  ld_async_f32(lds_dst + 0, gsrc + 0);
  ld_async_f32(lds_dst + 1, gsrc + 1);
  ld_async_f32(lds_dst + 2, gsrc + 2);
  ld_async_f32(lds_dst + 3, gsrc + 3);
#endif
}

__device__ __forceinline__ void wait_async0() {
#if __has_builtin(__builtin_amdgcn_s_wait_asynccnt)
  __builtin_amdgcn_s_wait_asynccnt(0);
#else
  asm volatile("s_wait_asynccnt 0" ::: "memory");
#endif
}

// 5-tap 4th-order Laplacian taps, K[d], zero outside [0,4]
__device__ __forceinline__ float kcoef(int d) {
  constexpr float C4 = (float)((0.001 / 5.0) * (0.001 / 5.0) / 12.0);
  float r = 0.0f;
  r = (d == 0 || d == 4) ? -C4 : r;
  r = (d == 1 || d == 3) ? 16.0f * C4 : r;
  r = (d == 2) ? -30.0f * C4 : r;
  return r;
}

__global__ __launch_bounds__(NT) void Propagator1D_82016695484589_kernel(
    const float* __restrict__ vel,      // [NV, NZ]
    const float* __restrict__ wavelet,  // [NV]
    const int*   __restrict__ zs,       // [NV, NS]
    const int*   __restrict__ zr,       // [NV, NS]
    const float* __restrict__ prev,     // [NV, NS, LP]
    const float* __restrict__ curr,     // [NV, NS, LP]
    float* __restrict__ out_curr,       // [NV, NS, LP]   (copy of curr)
    float* __restrict__ out_next,       // [NV, NS, LP]   (u_next)
    float* __restrict__ out_yt,         // [NV, NS]
    float* __restrict__ out_reg)        // [1]
{
  // s_curr col c  <->  global z = z0 - 4 + c   (halo-4 so interior is 16B aligned)
  __shared__ float s_curr[16 * CPITCH];
  // s_pu col c    <->  global z = z0 + c       (prev in -> u_next out, in place)
  __shared__ float s_pu[16 * PITCH];
  __shared__ float s_v2[PITCH];

  const int tid = threadIdx.x;
  const int z0  = blockIdx.x * ZC;
  const int s0  = blockIdx.y * 16;
  const int v   = blockIdx.z;
  const long long rowbase = ((long long)v * NS + s0) * LP;

  if (blockIdx.x == 0 && blockIdx.y == 0 && v == 0 && tid == 0) out_reg[0] = 0.0f;

  // ---- stage vel^2 (bottom edge-padded, matching jnp.pad(mode="edge")) ----
  for (int i = tid; i < ZC; i += NT) {
    int zg = z0 + i;
    int zc = zg < NZv ? zg : NZv - 1;
    float vv = vel[(long long)v * NZv + zc];
    s_v2[i] = vv * vv;
  }
  // ---- stage curr (+-2 halo), async b32 (halo shift breaks b128 align) ----
  for (int i = tid; i < 16 * (ZC + 4); i += NT) {
    int row = i / (ZC + 4);
    int k   = i - row * (ZC + 4);        // staged index 0..ZC+3
    int zg  = z0 - 2 + k;
    float* dst = &s_curr[row * CPITCH + (k + 2)];
    if (zg >= 0 && zg < LP)
      ld_async_f32(dst, &curr[rowbase + (long long)row * LP + zg]);
    else
      *dst = 0.0f;                       // zero 'same' padding of the stencil
  }
  // ---- stage prev, async b128 (16B aligned on both sides) -----------------
  for (int i = tid; i < 16 * (ZC / 4); i += NT) {
    int row = i / (ZC / 4);
    int col = (i - row * (ZC / 4)) * 4;
    int zg  = z0 + col;                  // multiple of 4; LP % 4 == 0
    float* dst = &s_pu[row * PITCH + col];
    if (zg < LP)
      ld_async_f32x4(dst, &prev[rowbase + (long long)row * LP + zg]);
    else {
      dst[0] = 0.0f; dst[1] = 0.0f; dst[2] = 0.0f; dst[3] = 0.0f;
    }
  }
  wait_async0();
  __syncthreads();

  // ---- per-lane constants --------------------------------------------------
  const int lane = tid & 31;
  const int wv   = tid >> 5;
  const int n    = lane & 15;   // N (shot column) for B/C/D ; also M for A
  const int hi   = lane >> 4;   // half-wave selector (K pairing / M+8)
  const int sg   = s0 + n;
  const float wav = wavelet[v];
  const int zsn = zs[v * NS + sg];
  const int zrn = zr[v * NS + sg];

  // A fragments: A_c[m][t] = K[t - m], t = c*4 + k, per 16x4 f32 layout
  // (lanes 0-15: M=0-15, K={0,1}; lanes 16-31: M=0-15, K={2,3})
  v2f A[5];
#pragma unroll
  for (int c = 0; c < 5; ++c) {
    int t0 = c * 4 + 2 * hi;
    A[c].x = kcoef(t0 - n);
    A[c].y = kcoef(t0 + 1 - n);
  }

  // ---- 16z x 16s WMMA subtiles; wave w handles subtiles w, w+8, w+16 ------
  for (int st = wv; st < ZC / 16; st += 8) {
    const int zb = st * 16;               // subtile z start (chunk-local)
    if (z0 + zb >= LP) break;

    v8f acc = {};
#pragma unroll
    for (int c = 0; c < 5; ++c) {
      // B_c[k][n] = curr[z = z0 + zb + (c*4+k) - 2]; LDS col = zb + c*4+k + 2
      int t0 = zb + c * 4 + 2 * hi + 2;
      v2f b;
      b.x = s_curr[n * CPITCH + t0];
      b.y = s_curr[n * CPITCH + t0 + 1];
      acc = __builtin_amdgcn_wmma_f32_16x16x4_f32(
          false, A[c], false, b, (short)0, acc, false, false);
    }

    // epilogue: D VGPR r holds M = r + 8*hi, N = lane&15
#pragma unroll
    for (int r = 0; r < 8; ++r) {
      int m  = r + 8 * hi;
      int zl = zb + m;                    // chunk-local z
      int zg = z0 + zl;                   // global padded z
      float u = acc[r] * s_v2[zl]
              + 2.0f * s_curr[n * CPITCH + zl + 4]
              - s_pu[n * PITCH + zl];
      if (zg == zsn) u += wav;            // source injection (pre-mask)
      if (zg >= NZv) {                    // sponge: exp(-(0.0015*k)^2), k=z-3999
        float k = (float)(zg - (NZv - 1));
        u *= __expf(-2.25e-6f * k * k);
      }
      s_pu[n * PITCH + zl] = u;           // in-place: prev tile becomes u_next
      if (zg == zrn && zg < LP) out_yt[v * NS + sg] = u;  // receiver pick
    }
  }
  __syncthreads();

  // ---- b128 non-temporal streaming of both full-size outputs --------------
  for (int i = tid; i < 16 * (ZC / 4); i += NT) {
    int row = i / (ZC / 4);
    int col = (i - row * (ZC / 4)) * 4;
    int zg  = z0 + col;                  // multiple of 4; whole float4 in/out
    if (zg < LP) {
      long long g = rowbase + (long long)row * LP + zg;
      v4f un = *(const v4f*)&s_pu[row * PITCH + col];
      v4f cc = *(const v4f*)&s_curr[row * CPITCH + col + 4];
      __builtin_nontemporal_store(un, (v4f*)&out_next[g]);
      __builtin_nontemporal_store(cc, (v4f*)&out_curr[g]);
    }
  }
}

extern "C" void kernel_launch(void* const* d_in, const int* in_sizes, int n_in,
                              void* d_out, int out_size, void* d_ws, size_t ws_size,
                              hipStream_t stream) {
  const float* vel     = (const float*)d_in[0];
  const float* wavelet = (const float*)d_in[1];
  const int*   zs      = (const int*)d_in[2];
  const int*   zr      = (const int*)d_in[3];
  const float* prev    = (const float*)d_in[4];
  const float* curr    = (const float*)d_in[5];

  float* out = (float*)d_out;
  const long long N = (long long)NV * NS * LP;  // 33,587,200
  float* out_curr = out;                        // output 0: curr copy
  float* out_next = out + N;                    // output 1: u_next
  float* out_yt   = out + 2 * N;                // output 2: yt_pred [NV*NS]
  float* out_reg  = out + 2 * N + (long long)NV * NS;  // output 3: regularizer

  dim3 grid((LP + ZC - 1) / ZC, NS / 16, NV);   // (11, 4, 128)
  Propagator1D_82016695484589_kernel<<<grid, NT, 0, stream>>>(
      vel, wavelet, zs, zr, prev, curr, out_curr, out_next, out_yt, out_reg);
}